// BirmanSchwingerCore_23304492548728
// MI455X (gfx1250) — compile-verified
//
#include <hip/hip_runtime.h>
#include <math.h>

typedef __attribute__((ext_vector_type(2))) float v2f;
typedef __attribute__((ext_vector_type(8))) float v8f;

#define BS_N   4096
#define BS_NT  256      // N/16 tiles per dimension
#define BS_TPW 8        // tiles per wave along v
#define BS_WPB 8        // waves per 256-thread block

// Stage 1: AS[b,i] = a*sin(2i), AC[b,i] = a*cos(2i),
// a = sqrt(|V[b,i]|+1e-10) / (1+i)
__global__ void bs_precompute(const float* __restrict__ V,
                              float* __restrict__ AS,
                              float* __restrict__ AC,
                              int total) {
    int idx = blockIdx.x * blockDim.x + threadIdx.x;
    if (idx >= total) return;
    int i = idx & (BS_N - 1);
    float a = sqrtf(fabsf(V[idx]) + 1e-10f) / (1.0f + (float)i);
    float s, c;
    sincosf(2.0f * (float)i, &s, &c);
    AS[idx] = a * s;
    AC[idx] = a * c;
}

// Stage 2: one wave computes a strip of BS_TPW 16x16 complex tiles.
// Per tile: D_re = A(16x4) x Bsin(4x16), D_im = A x Bcos via V_WMMA_F32_16X16X4_F32.
//   A  cols: [AS_u, AC_u, 0, 0]
//   Bsin rows: [-0.5*s*AC_v ; +0.5*s*AS_v ; 0 ; 0]   -> D_re = -0.5*s*a_u*a_v*sin(2(u-v))
//   Bcos rows: [+0.5*s*AS_v ; +0.5*s*AC_v ; 0 ; 0]   -> D_im = +0.5*s*a_u*a_v*cos(2(u-v))
// s = sgn per tile (uniform off-diagonal); diagonal tiles fixed up per element.
__global__ __launch_bounds__(256) void bs_tiles(const float* __restrict__ AS,
                                                const float* __restrict__ AC,
                                                float* __restrict__ out,
                                                int B) {
    const int lane = threadIdx.x & 31;
    const int wave = threadIdx.x >> 5;
    const int half = lane >> 4;      // 0: lanes 0-15, 1: lanes 16-31
    const int l    = lane & 15;

    int gw     = blockIdx.x * BS_WPB + wave;     // global wave id
    int tu_row = gw >> 5;                        // b*BS_NT + tu  (32 strips per row)
    int strip  = gw & 31;
    int b      = tu_row >> 8;                    // / BS_NT
    int tu     = tu_row & (BS_NT - 1);
    if (b >= B) return;

    const int u0 = tu << 4;
    const float* ASb = AS + b * BS_N;
    const float* ACb = AC + b * BS_N;

    // A operand: VGPR0 = K0 = AS_u, VGPR1 = K1 = AC_u (lanes 0-15); upper half zero
    float au0 = ASb[u0 + l];
    float au1 = ACb[u0 + l];
    au0 = half ? 0.0f : au0;
    au1 = half ? 0.0f : au1;
    v2f Amat;
    Amat.x = au0;
    Amat.y = au1;

    // row base for this lane's half (rows u0+8*half .. +7 stride over j)
    size_t row_base = ((size_t)b * BS_N + (size_t)(u0 + 8 * half)) * (size_t)(BS_N * 2);

    for (int t = 0; t < BS_TPW; ++t) {
        int tv = strip * BS_TPW + t;
        int v0 = tv << 4;
        float s = (tu >= tv) ? 1.0f : -1.0f;   // diagonal tile uses +1, fixed up below

        float asv = ASb[v0 + l];
        float acv = ACb[v0 + l];
        asv = half ? 0.0f : asv;
        acv = half ? 0.0f : acv;

        v2f Bsin, Bcos;
        Bsin.x = -0.5f * s * acv;   // K0 row
        Bsin.y =  0.5f * s * asv;   // K1 row
        Bcos.x =  0.5f * s * asv;
        Bcos.y =  0.5f * s * acv;

        v8f zero = {};
        v8f Dre = __builtin_amdgcn_wmma_f32_16x16x4_f32(
            false, Amat, false, Bsin, (short)0, zero, false, false);
        v8f Dim = __builtin_amdgcn_wmma_f32_16x16x4_f32(
            false, Amat, false, Bcos, (short)0, zero, false, false);

        if (tu == tv) {
            // per-element sgn(u-v): VGPR j holds row u_local = j + 8*half, col l
#pragma unroll
            for (int j = 0; j < 8; ++j) {
                int ul = j + 8 * half;
                float sg = (ul > l) ? 1.0f : ((ul < l) ? -1.0f : 0.0f);
                Dre[j] *= sg;
                Dim[j] *= sg;
            }
        }

        float* colp = out + row_base + (size_t)((v0 + l) * 2);
#pragma unroll
        for (int j = 0; j < 8; ++j) {
            v2f val;
            val.x = Dre[j];
            val.y = Dim[j];
            __builtin_nontemporal_store(val,
                (v2f*)(colp + (size_t)j * (size_t)(BS_N * 2)));
        }
    }
}

extern "C" void kernel_launch(void* const* d_in, const int* in_sizes, int n_in,
                              void* d_out, int out_size, void* d_ws, size_t ws_size,
                              hipStream_t stream) {
    (void)n_in; (void)out_size; (void)ws_size;
    const float* V = (const float*)d_in[0];
    float* out = (float*)d_out;
    const int total = in_sizes[0];          // B * N
    const int B = total / BS_N;

    float* AS = (float*)d_ws;               // B*N floats
    float* AC = AS + total;                 // B*N floats  (128 KB total for B=4)

    bs_precompute<<<(total + 255) / 256, 256, 0, stream>>>(V, AS, AC, total);

    // waves = B * NT * (NT/TPW) = B * 8192 ; blocks = waves / WPB
    int blocks = B * (BS_NT * (BS_NT / BS_TPW)) / BS_WPB;   // B*1024
    bs_tiles<<<blocks, 256, 0, stream>>>(AS, AC, out, B);
}